// OODGAT_32126355374946
// MI455X (gfx1250) — compile-verified
//
#include <hip/hip_runtime.h>

typedef __attribute__((ext_vector_type(2))) float v2f;
typedef __attribute__((ext_vector_type(8))) float v8f;

#define NHEADS 4
#define TPB 256

// ---------------------------------------------------------------------------
// C[M,NC] = A[M,K] * B[K,NC], row-major f32, via V_WMMA_F32_16X16X4_F32.
// Requires M%16==0, K%4==0, NC%16==0 (true here: M=50000, K in {256,128},
// NC in {128,160}). One wave per 16x16 output tile.
// A-frag (16x4):  lane(0..15)=row M, vgpr0=A[m][k+2*hi], vgpr1=A[m][k+2*hi+1]
// B-frag (4x16):  lane(0..15)=col N, vgpr0=B[k+2*hi][n], vgpr1=B[k+2*hi+1][n]
// C/D (16x16):    vgpr r -> M=r+8*hi, N=lane&15
// ---------------------------------------------------------------------------
__global__ __launch_bounds__(TPB) void gemm_f32_wmma(
    const float* __restrict__ A, const float* __restrict__ B,
    float* __restrict__ C, int M, int K, int NC) {
  const int tiles_n = NC >> 4;
  const int total = (M >> 4) * tiles_n;
  int wave = (int)((blockIdx.x * (unsigned)blockDim.x + threadIdx.x) >> 5);
  if (wave >= total) return;
  int tm = wave / tiles_n;
  int tn = wave - tm * tiles_n;
  int lane = threadIdx.x & 31;
  int lo = lane & 15, hi = lane >> 4;

  const float* arow = A + (size_t)(tm * 16 + lo) * K + 2 * hi;
  const float* bcol = B + (size_t)(2 * hi) * NC + tn * 16 + lo;

  v8f acc = {};
  for (int k = 0; k < K; k += 4) {
    v2f a, b;
    a[0] = arow[k];
    a[1] = arow[k + 1];
    b[0] = bcol[(size_t)k * NC];
    b[1] = bcol[(size_t)k * NC + NC];
    acc = __builtin_amdgcn_wmma_f32_16x16x4_f32(false, a, false, b,
                                                (short)0, acc, false, false);
  }
  float* crow = C + (size_t)(tm * 16 + 8 * hi) * NC + tn * 16 + lo;
#pragma unroll
  for (int r = 0; r < 8; ++r) crow[(size_t)r * NC] = acc[r];
}

// ---------------------------------------------------------------------------
// Per (node, head): alpha = <xh[n,h,:], att_q[h,:]>, sa = sigmoid(alpha).
// Also initializes the segment-max (encoded -inf) and segment-sum buffers.
// ---------------------------------------------------------------------------
__global__ void node_alpha_init(const float* __restrict__ xh,
                                const float* __restrict__ att_q,
                                float* __restrict__ sa,
                                unsigned* __restrict__ mmax,
                                float* __restrict__ ssum, int N, int C) {
  int i = blockIdx.x * blockDim.x + threadIdx.x;
  if (i >= N * NHEADS) return;
  int n = i >> 2, h = i & 3;
  const float* row = xh + (size_t)n * NHEADS * C + (size_t)h * C;
  const float* q = att_q + h * C;
  float a = 0.f;
  for (int c = 0; c < C; ++c) a += row[c] * q[c];
  sa[i] = 1.0f / (1.0f + expf(-a));
  mmax[i] = 0x007FFFFFu;  // order-preserving encoding of -inf (~bits(-inf))
  ssum[i] = 0.f;
}

__global__ void zero_f32(float* __restrict__ p, long long n) {
  long long i = blockIdx.x * (long long)blockDim.x + threadIdx.x;
  if (i < n) p[i] = 0.f;
}

__device__ __forceinline__ unsigned enc_f32(float f) {
  unsigned u = __float_as_uint(f);
  return (u & 0x80000000u) ? ~u : (u | 0x80000000u);
}
__device__ __forceinline__ float dec_f32(unsigned u) {
  unsigned b = (u & 0x80000000u) ? (u ^ 0x80000000u) : ~u;
  return __uint_as_float(b);
}

// ---------------------------------------------------------------------------
// Pass 1 over edges: e = w_alpha * w_beta, store + segment max over dst.
// ---------------------------------------------------------------------------
__global__ void edge_score(const float* __restrict__ xh,
                           const float* __restrict__ att_v,
                           const float* __restrict__ sa,
                           const int* __restrict__ src,
                           const int* __restrict__ dst,
                           float* __restrict__ ebuf,
                           unsigned* __restrict__ mmax, int E, int C) {
  int i = blockIdx.x * blockDim.x + threadIdx.x;
  if (i >= E * NHEADS) return;
  int e = i >> 2, h = i & 3;
  int s = src[e], d = dst[e];
  const float* xs = xh + (size_t)s * NHEADS * C + (size_t)h * C;
  const float* xd = xh + (size_t)d * NHEADS * C + (size_t)h * C;
  const float* v = att_v + h * C;
  float wb = 0.f;
  for (int c = 0; c < C; ++c) {
    float z = xd[c] + xs[c];
    z = (z > 0.f) ? z : 0.01f * z;  // leaky_relu, slope 0.01
    wb += v[c] * z;
  }
  float wa = 1.0f - fabsf(sa[d * NHEADS + h] - sa[s * NHEADS + h]);
  float ev = wa * wb;
  ebuf[i] = ev;
  atomicMax(&mmax[d * NHEADS + h], enc_f32(ev));
}

// ---------------------------------------------------------------------------
// Pass 2 over edges: ex = exp(e - max[dst]); segment sum over dst.
// ---------------------------------------------------------------------------
__global__ void edge_exp_sum(const int* __restrict__ dst,
                             float* __restrict__ ebuf,
                             const unsigned* __restrict__ mmax,
                             float* __restrict__ ssum, int E) {
  int i = blockIdx.x * blockDim.x + threadIdx.x;
  if (i >= E * NHEADS) return;
  int e = i >> 2, h = i & 3;
  int d = dst[e];
  float mx = dec_f32(mmax[d * NHEADS + h]);
  float ex = expf(ebuf[i] - mx);
  ebuf[i] = ex;
  atomicAdd(&ssum[d * NHEADS + h], ex);
}

// ---------------------------------------------------------------------------
// Pass 3: out[dst] += xh[src] * w, one thread per (edge, h*C+c) element.
// ---------------------------------------------------------------------------
__global__ void edge_scatter(const float* __restrict__ xh,
                             const int* __restrict__ src,
                             const int* __restrict__ dst,
                             const float* __restrict__ ebuf,
                             const float* __restrict__ ssum,
                             float* __restrict__ acc, int E, int C) {
  long long i = blockIdx.x * (long long)blockDim.x + threadIdx.x;
  int HC = NHEADS * C;
  if (i >= (long long)E * HC) return;
  int e = (int)(i / HC);
  int j = (int)(i - (long long)e * HC);
  int h = j / C;
  int s = src[e], d = dst[e];
  float w = ebuf[e * NHEADS + h] / (ssum[d * NHEADS + h] + 1e-16f);
  atomicAdd(&acc[(size_t)d * HC + j], xh[(size_t)s * HC + j] * w);
}

__global__ void bias_elu_inplace(float* __restrict__ a,
                                 const float* __restrict__ bias,
                                 long long total, int F) {
  long long i = blockIdx.x * (long long)blockDim.x + threadIdx.x;
  if (i >= total) return;
  int j = (int)(i % F);
  float v = a[i] + bias[j];
  a[i] = (v > 0.f) ? v : (expf(v) - 1.f);
}

__global__ void mean_heads_bias(const float* __restrict__ acc,
                                const float* __restrict__ bias,
                                float* __restrict__ out, int N, int C) {
  int i = blockIdx.x * blockDim.x + threadIdx.x;
  if (i >= N * C) return;
  int n = i / C, c = i - n * C;
  const float* p = acc + (size_t)n * NHEADS * C + c;
  float v = 0.f;
#pragma unroll
  for (int h = 0; h < NHEADS; ++h) v += p[h * C];
  out[i] = v * (1.0f / NHEADS) + bias[c];
}

// ---------------------------------------------------------------------------
extern "C" void kernel_launch(void* const* d_in, const int* in_sizes, int n_in,
                              void* d_out, int out_size, void* d_ws,
                              size_t ws_size, hipStream_t stream) {
  const float* x      = (const float*)d_in[0];
  const int*   src    = (const int*)d_in[1];
  const int*   dstv   = (const int*)d_in[2];
  const float* lin1   = (const float*)d_in[3];
  const float* att_q1 = (const float*)d_in[4];
  const float* att_v1 = (const float*)d_in[5];
  const float* bias1  = (const float*)d_in[6];
  const float* lin2   = (const float*)d_in[7];
  const float* att_q2 = (const float*)d_in[8];
  const float* att_v2 = (const float*)d_in[9];
  const float* bias2  = (const float*)d_in[10];

  const int IN_DIM = 256, C1 = 32, C2 = 40;
  const int F1 = NHEADS * C1;  // 128
  const int F2 = NHEADS * C2;  // 160
  const int N = in_sizes[0] / IN_DIM;  // 50000 (multiple of 16)
  const int E = in_sizes[1];           // 850000

  char* ws = (char*)d_ws;
  size_t off = 0;
  auto alloc = [&](size_t bytes) -> void* {
    void* p = ws + off;
    off += (bytes + 255) & ~(size_t)255;
    return p;
  };
  float*    xh1  = (float*)alloc((size_t)N * F1 * 4);      // 25.6 MB
  float*    sa   = (float*)alloc((size_t)N * NHEADS * 4);  // 0.8 MB (reused)
  unsigned* mmax = (unsigned*)alloc((size_t)N * NHEADS * 4);
  float*    ssum = (float*)alloc((size_t)N * NHEADS * 4);
  float*    ebuf = (float*)alloc((size_t)E * NHEADS * 4);  // 13.6 MB (reused)
  float*    acc1 = (float*)alloc((size_t)N * F1 * 4);      // becomes h (ELU in-place)
  float*    xh2  = (float*)alloc((size_t)N * F2 * 4);      // 32 MB
  float*    acc2 = (float*)alloc((size_t)N * F2 * 4);      // 32 MB
  (void)ws_size; (void)n_in; (void)out_size;

  auto blocks = [](long long work) -> unsigned {
    return (unsigned)((work + TPB - 1) / TPB);
  };

  // ---------------- layer 1 ----------------
  {
    long long waves = (long long)(N / 16) * (F1 / 16);
    gemm_f32_wmma<<<blocks(waves * 32), TPB, 0, stream>>>(x, lin1, xh1, N, IN_DIM, F1);
  }
  node_alpha_init<<<blocks((long long)N * NHEADS), TPB, 0, stream>>>(
      xh1, att_q1, sa, mmax, ssum, N, C1);
  zero_f32<<<blocks((long long)N * F1), TPB, 0, stream>>>(acc1, (long long)N * F1);
  edge_score<<<blocks((long long)E * NHEADS), TPB, 0, stream>>>(
      xh1, att_v1, sa, src, dstv, ebuf, mmax, E, C1);
  edge_exp_sum<<<blocks((long long)E * NHEADS), TPB, 0, stream>>>(
      dstv, ebuf, mmax, ssum, E);
  edge_scatter<<<blocks((long long)E * F1), TPB, 0, stream>>>(
      xh1, src, dstv, ebuf, ssum, acc1, E, C1);
  bias_elu_inplace<<<blocks((long long)N * F1), TPB, 0, stream>>>(
      acc1, bias1, (long long)N * F1, F1);

  // ---------------- layer 2 ----------------
  {
    long long waves = (long long)(N / 16) * (F2 / 16);
    gemm_f32_wmma<<<blocks(waves * 32), TPB, 0, stream>>>(acc1, lin2, xh2, N, F1, F2);
  }
  node_alpha_init<<<blocks((long long)N * NHEADS), TPB, 0, stream>>>(
      xh2, att_q2, sa, mmax, ssum, N, C2);
  zero_f32<<<blocks((long long)N * F2), TPB, 0, stream>>>(acc2, (long long)N * F2);
  edge_score<<<blocks((long long)E * NHEADS), TPB, 0, stream>>>(
      xh2, att_v2, sa, src, dstv, ebuf, mmax, E, C2);
  edge_exp_sum<<<blocks((long long)E * NHEADS), TPB, 0, stream>>>(
      dstv, ebuf, mmax, ssum, E);
  edge_scatter<<<blocks((long long)E * F2), TPB, 0, stream>>>(
      xh2, src, dstv, ebuf, ssum, acc2, E, C2);
  mean_heads_bias<<<blocks((long long)N * C2), TPB, 0, stream>>>(
      acc2, bias2, (float*)d_out, N, C2);
}